// GATv2Regressor_395136991715
// MI455X (gfx1250) — compile-verified
//
#include <hip/hip_runtime.h>

// ---------------------------------------------------------------------------
// GATv2 regressor for MI455X (gfx1250, wave32).
// GEMMs use V_WMMA_F32_16X16X4_F32 (native f32 matrix path - reference is f32
// and GEMMs are not the bottleneck, so no precision downcast needed).
// Edge phase: wave-per-edge gather + shuffle-reduced attention dot +
// atomic segment-max / segment-sum softmax (working set fits in 192MB L2).
// ---------------------------------------------------------------------------

typedef __attribute__((ext_vector_type(2))) float v2f;
typedef __attribute__((ext_vector_type(8))) float v8f;

#define NEG_SLOPE 0.2f

__device__ __forceinline__ unsigned fkey(float f) {
  unsigned u = __float_as_uint(f);
  return (u >> 31) ? ~u : (u | 0x80000000u);   // monotone order-preserving key
}
__device__ __forceinline__ float kfloat(unsigned k) {
  unsigned u = (k & 0x80000000u) ? (k & 0x7fffffffu) : ~k;
  return __uint_as_float(u);
}
__device__ __forceinline__ float lrelu(float v) { return v > 0.f ? v : NEG_SLOPE * v; }

// -------------------------------- utility ----------------------------------
__global__ void zero_kernel(unsigned* __restrict__ p, long long n) {
  long long i = (long long)blockIdx.x * blockDim.x + threadIdx.x;
  long long stride = (long long)gridDim.x * blockDim.x;
  for (; i < n; i += stride) p[i] = 0u;
}

// ------------------------------- WMMA GEMM ---------------------------------
// C[M,NDIM] = A[M,KDIM] @ W[KDIM,NDIM] + bias (optional relu).
// One wave computes one 16x16 tile; K advances 4/instruction (f32 WMMA).
// A frag (16x4): lanes 0-15 rows, VGPR0/1 = K {0,1} (halves 16-31: K {2,3}).
// B frag (4x16): lanes hold column tN+(lane&15), K pair per half.
template <int KDIM, int NDIM, bool RELU>
__global__ __launch_bounds__(128) void wmma_gemm_f32(
    const float* __restrict__ A, const float* __restrict__ W,
    const float* __restrict__ bias, float* __restrict__ C, int M) {
  const int ntn = NDIM / 16;
  int tile = blockIdx.x * 4 + (threadIdx.x >> 5);
  int total = (M / 16) * ntn;
  if (tile >= total) return;               // wave-uniform: EXEC stays all-1s
  int tM = (tile / ntn) * 16;
  int tN = (tile % ntn) * 16;
  int lane = threadIdx.x & 31;
  int half = lane >> 4;
  int l = lane & 15;
  int arow = tM + l;
  int bcol = tN + l;
  v8f acc = {};
#pragma unroll
  for (int k = 0; k < KDIM; k += 4) {
    int kk = k + half * 2;
    float2 av = *(const float2*)&A[arow * KDIM + kk];   // 8B aligned (kk even)
    v2f a; a.x = av.x; a.y = av.y;
    v2f b; b.x = W[kk * NDIM + bcol]; b.y = W[(kk + 1) * NDIM + bcol];
    acc = __builtin_amdgcn_wmma_f32_16x16x4_f32(false, a, false, b, (short)0,
                                                acc, false, false);
  }
  float bv = bias[bcol];
#pragma unroll
  for (int v = 0; v < 8; ++v) {
    int row = tM + half * 8 + v;
    float r = acc[v] + bv;
    if (RELU) r = fmaxf(r, 0.f);
    C[row * NDIM + bcol] = r;
  }
}

// --------------------------- edge kernels (H=2) ----------------------------
// wave per edge; lane handles 4 features; lanes 0-15 -> head 0, 16-31 -> head 1
__global__ __launch_bounds__(256) void edge_score_h2(
    const float* __restrict__ xl, const float* __restrict__ xr,
    const float* __restrict__ att, const int* __restrict__ src,
    const int* __restrict__ dst, int E0, int ET, float* __restrict__ score,
    unsigned* __restrict__ mkey) {
  int e = blockIdx.x * 8 + (threadIdx.x >> 5);
  if (e >= ET) return;
  int lane = threadIdx.x & 31;
  int s, d;
  if (e < E0) { s = src[e]; d = dst[e]; } else { s = d = e - E0; }
  int f = lane * 4;
  float4 a = *(const float4*)&xl[s * 128 + f];
  float4 b = *(const float4*)&xr[d * 128 + f];
  float4 t = *(const float4*)&att[f];
  float p = lrelu(a.x + b.x) * t.x + lrelu(a.y + b.y) * t.y +
            lrelu(a.z + b.z) * t.z + lrelu(a.w + b.w) * t.w;
  p += __shfl_xor(p, 1, 32);
  p += __shfl_xor(p, 2, 32);
  p += __shfl_xor(p, 4, 32);
  p += __shfl_xor(p, 8, 32);               // reduce within each 16-lane head
  if ((lane & 15) == 0) {
    int h = lane >> 4;
    score[e * 2 + h] = p;
    atomicMax(&mkey[d * 2 + h], fkey(p));
  }
}

__global__ __launch_bounds__(256) void edge_aggr_h2(
    const float* __restrict__ xl, const float* __restrict__ score,
    const unsigned* __restrict__ mkey, const int* __restrict__ src,
    const int* __restrict__ dst, int E0, int ET, float* __restrict__ num,
    float* __restrict__ den) {
  int e = blockIdx.x * 8 + (threadIdx.x >> 5);
  if (e >= ET) return;
  int lane = threadIdx.x & 31;
  int s, d;
  if (e < E0) { s = src[e]; d = dst[e]; } else { s = d = e - E0; }
  int h = lane >> 4;
  float ex = __expf(score[e * 2 + h] - kfloat(mkey[d * 2 + h]));
  int f = lane * 4;
  float4 a = *(const float4*)&xl[s * 128 + f];
  float* np = &num[d * 128 + f];
  atomicAdd(np + 0, ex * a.x);
  atomicAdd(np + 1, ex * a.y);
  atomicAdd(np + 2, ex * a.z);
  atomicAdd(np + 3, ex * a.w);
  if ((lane & 15) == 0) atomicAdd(&den[d * 2 + h], ex);
}

// --------------------------- edge kernels (H=1) ----------------------------
__global__ __launch_bounds__(256) void edge_score_h1(
    const float* __restrict__ xl, const float* __restrict__ xr,
    const float* __restrict__ att, const int* __restrict__ src,
    const int* __restrict__ dst, int E0, int ET, float* __restrict__ score,
    unsigned* __restrict__ mkey) {
  int e = blockIdx.x * 8 + (threadIdx.x >> 5);
  if (e >= ET) return;
  int lane = threadIdx.x & 31;
  int s, d;
  if (e < E0) { s = src[e]; d = dst[e]; } else { s = d = e - E0; }
  int f = lane * 2;
  float2 a = *(const float2*)&xl[s * 64 + f];
  float2 b = *(const float2*)&xr[d * 64 + f];
  float2 t = *(const float2*)&att[f];
  float p = lrelu(a.x + b.x) * t.x + lrelu(a.y + b.y) * t.y;
  p += __shfl_xor(p, 1, 32);
  p += __shfl_xor(p, 2, 32);
  p += __shfl_xor(p, 4, 32);
  p += __shfl_xor(p, 8, 32);
  p += __shfl_xor(p, 16, 32);
  if (lane == 0) {
    score[e] = p;
    atomicMax(&mkey[d], fkey(p));
  }
}

__global__ __launch_bounds__(256) void edge_aggr_h1(
    const float* __restrict__ xl, const float* __restrict__ score,
    const unsigned* __restrict__ mkey, const int* __restrict__ src,
    const int* __restrict__ dst, int E0, int ET, float* __restrict__ num,
    float* __restrict__ den) {
  int e = blockIdx.x * 8 + (threadIdx.x >> 5);
  if (e >= ET) return;
  int lane = threadIdx.x & 31;
  int s, d;
  if (e < E0) { s = src[e]; d = dst[e]; } else { s = d = e - E0; }
  float ex = __expf(score[e] - kfloat(mkey[d]));
  int f = lane * 2;
  float2 a = *(const float2*)&xl[s * 64 + f];
  atomicAdd(&num[d * 64 + f + 0], ex * a.x);
  atomicAdd(&num[d * 64 + f + 1], ex * a.y);
  if (lane == 0) atomicAdd(&den[d], ex);
}

// out[n,f] = relu(num[n,f] / (den[n,h]+eps) + bias[f])   (in-place safe)
__global__ void finalize_nodes(const float* __restrict__ num,
                               const float* __restrict__ den,
                               const float* __restrict__ bias,
                               float* __restrict__ out, int N, int H, int D) {
  int F = H * D;
  int i = blockIdx.x * blockDim.x + threadIdx.x;
  if (i >= N * F) return;
  int f = i % F;
  int n = i / F;
  int h = f / D;
  float v = num[i] / (den[n * H + h] + 1e-16f) + bias[f];
  out[i] = fmaxf(v, 0.f);
}

// ------------------------------- pooling -----------------------------------
// gate[n] = gtmp[n,:]·gw2 + gb2 ; segment-max over batch
__global__ __launch_bounds__(256) void gate_kernel(
    const float* __restrict__ gtmp, const float* __restrict__ gw2,
    const float* __restrict__ gb2, const int* __restrict__ batch,
    float* __restrict__ gate, unsigned* __restrict__ mB, int N) {
  int n = blockIdx.x * 8 + (threadIdx.x >> 5);
  if (n >= N) return;
  int lane = threadIdx.x & 31;
  float2 g = *(const float2*)&gtmp[n * 64 + lane * 2];
  float2 w = *(const float2*)&gw2[lane * 2];
  float p = g.x * w.x + g.y * w.y;
  p += __shfl_xor(p, 1, 32);
  p += __shfl_xor(p, 2, 32);
  p += __shfl_xor(p, 4, 32);
  p += __shfl_xor(p, 8, 32);
  p += __shfl_xor(p, 16, 32);
  if (lane == 0) {
    float v = p + gb2[0];
    gate[n] = v;
    atomicMax(&mB[batch[n]], fkey(v));
  }
}

__global__ __launch_bounds__(256) void pool_aggr(
    const float* __restrict__ h2, const float* __restrict__ gate,
    const unsigned* __restrict__ mB, const int* __restrict__ batch,
    float* __restrict__ numB, float* __restrict__ sB, int N) {
  int n = blockIdx.x * 8 + (threadIdx.x >> 5);
  if (n >= N) return;
  int lane = threadIdx.x & 31;
  int b = batch[n];
  float ex = __expf(gate[n] - kfloat(mB[b]));
  float2 hv = *(const float2*)&h2[n * 64 + lane * 2];
  atomicAdd(&numB[b * 64 + lane * 2 + 0], ex * hv.x);
  atomicAdd(&numB[b * 64 + lane * 2 + 1], ex * hv.y);
  if (lane == 0) atomicAdd(&sB[b], ex);
}

// final MLP head over B=64 pooled vectors; one thread per batch element
__global__ void head_kernel(const float* __restrict__ numB,
                            const float* __restrict__ sB,
                            const float* __restrict__ l1w,
                            const float* __restrict__ l1b,
                            const float* __restrict__ l2w,
                            const float* __restrict__ l2b,
                            float* __restrict__ out) {
  int b = threadIdx.x;  // 64 threads
  float pooled[64];
  float s = sB[b] + 1e-16f;
#pragma unroll
  for (int d = 0; d < 64; ++d) pooled[d] = numB[b * 64 + d] / s;
  float acc = 0.f;
  for (int j = 0; j < 64; ++j) {
    float t = l1b[j];
#pragma unroll
    for (int d = 0; d < 64; ++d) t += pooled[d] * l1w[d * 64 + j];
    acc += fmaxf(t, 0.f) * l2w[j];
  }
  out[b] = acc + l2b[0];
}

// ------------------------------- launcher ----------------------------------
extern "C" void kernel_launch(void* const* d_in, const int* in_sizes, int n_in,
                              void* d_out, int out_size, void* d_ws,
                              size_t ws_size, hipStream_t stream) {
  (void)n_in; (void)out_size; (void)ws_size;
  const float* x     = (const float*)d_in[0];
  const int*   ei    = (const int*)d_in[1];
  const int*   batch = (const int*)d_in[2];
  const float* w1_l  = (const float*)d_in[3];
  const float* b1_l  = (const float*)d_in[4];
  const float* w1_r  = (const float*)d_in[5];
  const float* b1_r  = (const float*)d_in[6];
  const float* att1  = (const float*)d_in[7];
  const float* bias1 = (const float*)d_in[8];
  const float* w2_l  = (const float*)d_in[9];
  const float* b2_l  = (const float*)d_in[10];
  const float* w2_r  = (const float*)d_in[11];
  const float* b2_r  = (const float*)d_in[12];
  const float* att2  = (const float*)d_in[13];
  const float* bias2 = (const float*)d_in[14];
  const float* gw1   = (const float*)d_in[15];
  const float* gb1   = (const float*)d_in[16];
  const float* gw2   = (const float*)d_in[17];
  const float* gb2   = (const float*)d_in[18];
  const float* l1w   = (const float*)d_in[19];
  const float* l1b   = (const float*)d_in[20];
  const float* l2w   = (const float*)d_in[21];
  const float* l2b   = (const float*)d_in[22];
  float* out = (float*)d_out;

  const int N = in_sizes[0] / 128;   // 50000 (divisible by 16)
  const int E = in_sizes[1] / 2;     // 800000
  const int ET = E + N;              // edges + self loops
  const int* src = ei;
  const int* dst = ei + E;

  // ---- workspace layout (all 4-byte words; reuse across phases) ----
  float* ws = (float*)d_ws;
  size_t o = 0;
  float* R0 = ws + o; o += (size_t)N * 128;          // xl1 ; later xl2|xr2
  float* R1 = ws + o; o += (size_t)N * 128;          // xr1 ; later gtmp|gate
  float* SC = ws + o; o += (size_t)ET * 2;           // edge scores
  unsigned* MK = (unsigned*)(ws + o); o += (size_t)N * 2;  // segment-max keys
  float* DEN = ws + o; o += (size_t)N * 2;           // softmax denominators
  float* R5 = ws + o; o += (size_t)N * 128;          // num/h1 ; later num2/h2
  float* TAIL = ws + o; o += 64 + 64 + 64 * 64;      // mB | sB | numB
  unsigned* mB = (unsigned*)TAIL;
  float* sB = TAIL + 64;
  float* numB = TAIL + 128;

  const int t1 = (N / 16) * (128 / 16);
  const int g1 = (t1 + 3) / 4;          // 4 waves (tiles) per 128-thread block
  const int t2 = (N / 16) * (64 / 16);
  const int g2 = (t2 + 3) / 4;
  const int edge_blocks = (ET + 7) / 8; // 8 waves (edges) per 256-thread block
  const int node_blocks = (N + 7) / 8;

  // ---------------- Layer 1 (heads=2, dim=64) ----------------
  wmma_gemm_f32<128, 128, false><<<g1, 128, 0, stream>>>(x, w1_l, b1_l, R0, N);
  wmma_gemm_f32<128, 128, false><<<g1, 128, 0, stream>>>(x, w1_r, b1_r, R1, N);
  zero_kernel<<<2048, 256, 0, stream>>>(MK, (long long)N * 132); // MK|DEN|R5
  edge_score_h2<<<edge_blocks, 256, 0, stream>>>(R0, R1, att1, src, dst, E, ET, SC, MK);
  edge_aggr_h2<<<edge_blocks, 256, 0, stream>>>(R0, SC, MK, src, dst, E, ET, R5, DEN);
  finalize_nodes<<<(N * 128 + 255) / 256, 256, 0, stream>>>(R5, DEN, bias1, R5, N, 2, 64);

  // ---------------- Layer 2 (heads=1, dim=64) ----------------
  wmma_gemm_f32<128, 64, false><<<g2, 128, 0, stream>>>(R5, w2_l, b2_l, R0, N);
  wmma_gemm_f32<128, 64, false><<<g2, 128, 0, stream>>>(R5, w2_r, b2_r, R0 + (size_t)N * 64, N);
  zero_kernel<<<2048, 256, 0, stream>>>(MK, (long long)N * 132);
  edge_score_h1<<<edge_blocks, 256, 0, stream>>>(R0, R0 + (size_t)N * 64, att2, src, dst, E, ET, SC, MK);
  edge_aggr_h1<<<edge_blocks, 256, 0, stream>>>(R0, SC, MK, src, dst, E, ET, R5, DEN);
  finalize_nodes<<<(N * 64 + 255) / 256, 256, 0, stream>>>(R5, DEN, bias2, R5, N, 1, 64);

  // ---------------- Global attention pool + MLP head ----------------
  wmma_gemm_f32<64, 64, true><<<g2, 128, 0, stream>>>(R5, gw1, gb1, R1, N);
  zero_kernel<<<32, 256, 0, stream>>>(mB, 64 + 64 + 64 * 64);
  gate_kernel<<<node_blocks, 256, 0, stream>>>(R1, gw2, gb2, batch, R1 + (size_t)N * 64, mB, N);
  pool_aggr<<<node_blocks, 256, 0, stream>>>(R5, R1 + (size_t)N * 64, mB, batch, numB, sB, N);
  head_kernel<<<1, 64, 0, stream>>>(numB, sB, l1w, l1b, l2w, l2b, out);
}